// DecoderModel_13426067767654
// MI455X (gfx1250) — compile-verified
//
#include <hip/hip_runtime.h>
#include <hip/hip_bf16.h>
#include <math.h>

#define BB 2
#define DD 64
#define LL 8192
#define CROSSC 2048
#define CINQ 2112
#define NCLS 48
#define TEMBD 512
#define KW 5
#define NLAYERS 8

typedef float v2f __attribute__((ext_vector_type(2)));
typedef float v8f __attribute__((ext_vector_type(8)));

__device__ __forceinline__ v8f wmma_f32(v2f a, v2f b, v8f c) {
  // D = A(16x4 f32) * B(4x16 f32) + C(16x16 f32)
  return __builtin_amdgcn_wmma_f32_16x16x4_f32(false, a, false, b, (short)0, c,
                                               false, false);
}

__device__ __forceinline__ float swishf(float x) {
  return x * (1.0f / (1.0f + expf(-x)));
}

// ---------------- time embedding MLP (tiny) ----------------
__global__ void time_embed_kernel(const int* __restrict__ t,
                                  const float* __restrict__ w1, const float* __restrict__ b1,
                                  const float* __restrict__ w2, const float* __restrict__ b2,
                                  const float* __restrict__ pw, const float* __restrict__ pb,
                                  float* __restrict__ tvec) {
  __shared__ float sA[TEMBD];
  __shared__ float sB[TEMBD];
  const int n = blockIdx.x;
  const int i = threadIdx.x;
  const float tv = (float)t[n];
  {
    const float neg = -logf(10000.0f) / 255.0f;
    int j = (i < 256) ? i : (i - 256);
    float ang = tv * expf((float)j * neg);
    sA[i] = (i < 256) ? sinf(ang) : cosf(ang);
  }
  __syncthreads();
  {
    float acc = b1[i];
    const float* wr = w1 + (size_t)i * TEMBD;
    for (int j = 0; j < TEMBD; ++j) acc += wr[j] * sA[j];
    sB[i] = swishf(acc);
  }
  __syncthreads();
  {
    float acc = b2[i];
    const float* wr = w2 + (size_t)i * TEMBD;
    for (int j = 0; j < TEMBD; ++j) acc += wr[j] * sB[j];
    sA[i] = swishf(acc);  // swish applied before time projection
  }
  __syncthreads();
  if (i < DD) {
    float acc = pb[i];
    const float* wr = pw + (size_t)i * TEMBD;
    for (int j = 0; j < TEMBD; ++j) acc += wr[j] * sA[j];
    tvec[n * DD + i] = acc;
  }
}

// ---------------- conv_in (48 -> 64, 1x1) + time bias ----------------
__global__ void conv_in_kernel(const float* __restrict__ w, const float* __restrict__ b,
                               const float* __restrict__ event, const float* __restrict__ tvec,
                               float* __restrict__ fra) {
  const int l = blockIdx.x * blockDim.x + threadIdx.x;
  const int d = blockIdx.y, n = blockIdx.z;
  float acc = b[d] + tvec[n * DD + d];
  const float* wr = w + d * NCLS;
  const float* ev = event + (size_t)n * NCLS * LL + l;
  #pragma unroll 8
  for (int c = 0; c < NCLS; ++c) acc += wr[c] * ev[(size_t)c * LL];
  fra[((size_t)n * DD + d) * LL + l] = acc;
}

// ---------------- q/k projection: [64x2112] x [2112x8192], fp32 WMMA ----------------
// 8 waves/block, each wave owns a 64(M) x 32(N) output tile for BOTH q and k.
// A-tiles come from LDS (staged once per block per K-chunk), every A LDS read
// feeds 2 WMMAs (two N tiles); every B global read feeds 8 WMMAs (4 M tiles x {q,k}).
__global__ void __launch_bounds__(256) qkproj_kernel(
    const float* __restrict__ qw, const float* __restrict__ qb,
    const float* __restrict__ kw, const float* __restrict__ kb,
    const float* __restrict__ fra, const float* __restrict__ xcross,
    float* __restrict__ xq, float* __restrict__ xk) {
  __shared__ float sQ[DD * 64];
  __shared__ float sK[DD * 64];
  const int tid = threadIdx.x;
  const int wave = tid >> 5, lane = tid & 31;
  const int n = blockIdx.y;
  const int l0 = blockIdx.x * 256 + wave * 32;
  const int lm = lane & 15;
  const int khalf = (lane < 16) ? 0 : 2;
  v8f accq[8], acck[8];  // [nt*4 + mt]
  v8f zero = {0.f, 0.f, 0.f, 0.f, 0.f, 0.f, 0.f, 0.f};
  #pragma unroll
  for (int a = 0; a < 8; ++a) { accq[a] = zero; acck[a] = zero; }

  const int nchunks = CINQ / 64;  // 33
  for (int kc = 0; kc < nchunks; ++kc) {
    // stage this K-chunk of both weight matrices into LDS (shared by 8 waves)
    for (int idx = tid; idx < DD * 64; idx += 256) {
      int d = idx >> 6, kkk = idx & 63;
      sQ[idx] = qw[(size_t)d * CINQ + kc * 64 + kkk];
      sK[idx] = kw[(size_t)d * CINQ + kc * 64 + kkk];
    }
    // prefetch next chunk's B rows (L2-resident x_cross) into near cache while
    // we are stalled on the staging barrier anyway
    if (kc + 1 < nchunks && tid < DD) {
      const float* nsrc = xcross + ((size_t)n * CROSSC + (size_t)kc * 64) * LL;
      __builtin_prefetch(nsrc + (size_t)tid * LL + blockIdx.x * 256, 0, 3);
    }
    __syncthreads();
    // cat(fra, x_cross): chunk 0 is fra channels, chunks 1..32 are x_cross
    const float* src = (kc == 0)
        ? (fra + (size_t)n * DD * LL)
        : (xcross + ((size_t)n * CROSSC + (size_t)(kc - 1) * 64) * LL);
    const int pos0 = l0 + lm;
    const int pos1 = pos0 + 16;
    for (int ks = 0; ks < 16; ++ks) {
      const int kk = ks * 4 + khalf;
      v2f bv0, bv1;
      bv0.x = src[(size_t)kk * LL + pos0];
      bv0.y = src[(size_t)(kk + 1) * LL + pos0];
      bv1.x = src[(size_t)kk * LL + pos1];
      bv1.y = src[(size_t)(kk + 1) * LL + pos1];
      #pragma unroll
      for (int mt = 0; mt < 4; ++mt) {
        const int row = mt * 16 + lm;
        v2f aq, ak;
        aq.x = sQ[row * 64 + kk]; aq.y = sQ[row * 64 + kk + 1];
        ak.x = sK[row * 64 + kk]; ak.y = sK[row * 64 + kk + 1];
        accq[mt]     = wmma_f32(aq, bv0, accq[mt]);
        accq[4 + mt] = wmma_f32(aq, bv1, accq[4 + mt]);
        acck[mt]     = wmma_f32(ak, bv0, acck[mt]);
        acck[4 + mt] = wmma_f32(ak, bv1, acck[4 + mt]);
      }
    }
    __syncthreads();
  }
  const int mhalf = (lane < 16) ? 0 : 8;
  #pragma unroll
  for (int nt = 0; nt < 2; ++nt)
    #pragma unroll
    for (int mt = 0; mt < 4; ++mt)
      #pragma unroll
      for (int j = 0; j < 8; ++j) {
        const int m = mt * 16 + j + mhalf;
        const size_t o = ((size_t)n * DD + m) * LL + (l0 + nt * 16 + lm);
        xq[o] = accq[nt * 4 + mt][j] + qb[m];
        xk[o] = acck[nt * 4 + mt][j] + kb[m];
      }
}

// ---------------- generic [Mout x 64] x [64 x 8192] WMMA GEMM ----------------
template <int MT, bool RELU, bool RES>
__global__ void __launch_bounds__(256) gemm64_kernel(
    const float* __restrict__ W, const float* __restrict__ bias,
    const float* __restrict__ In, const float* __restrict__ Res,
    float* __restrict__ Out) {
  __shared__ float sW[DD * DD];
  const int tid = threadIdx.x;
  const int wave = tid >> 5, lane = tid & 31;
  const int n = blockIdx.y;
  for (int idx = tid; idx < MT * 16 * 64; idx += 256) sW[idx] = W[idx];
  __syncthreads();
  const int l0 = blockIdx.x * 128 + wave * 16;
  const int lm = lane & 15;
  const int khalf = (lane < 16) ? 0 : 2;
  const float* in = In + (size_t)n * DD * LL;
  v8f acc[MT];
  v8f zero = {0.f, 0.f, 0.f, 0.f, 0.f, 0.f, 0.f, 0.f};
  #pragma unroll
  for (int mt = 0; mt < MT; ++mt) acc[mt] = zero;
  const int pos = l0 + lm;
  for (int ks = 0; ks < 16; ++ks) {
    const int kk = ks * 4 + khalf;
    v2f bv;
    bv.x = in[(size_t)kk * LL + pos];
    bv.y = in[(size_t)(kk + 1) * LL + pos];
    #pragma unroll
    for (int mt = 0; mt < MT; ++mt) {
      const int row = mt * 16 + lm;
      v2f av;
      av.x = sW[row * 64 + kk]; av.y = sW[row * 64 + kk + 1];
      acc[mt] = wmma_f32(av, bv, acc[mt]);
    }
  }
  const int mhalf = (lane < 16) ? 0 : 8;
  #pragma unroll
  for (int mt = 0; mt < MT; ++mt)
    #pragma unroll
    for (int j = 0; j < 8; ++j) {
      const int m = mt * 16 + j + mhalf;
      float v = acc[mt][j] + bias[m];
      if (RELU) v = fmaxf(v, 0.0f);
      if (RES) v += Res[((size_t)n * DD + m) * LL + (l0 + lm)];
      Out[((size_t)n * (MT * 16) + m) * LL + (l0 + lm)] = v;
    }
}

// ---------------- dilated K=5 conv (weights uniform per block -> scalar loads) ----------------
__global__ void dilconv_kernel(const float* __restrict__ cw, const float* __restrict__ cb,
                               const float* __restrict__ fra, float* __restrict__ s, int dil) {
  const int l = blockIdx.x * blockDim.x + threadIdx.x;
  const int d = blockIdx.y, n = blockIdx.z;
  float acc = cb[d];
  for (int c = 0; c < DD; ++c) {
    const float* base = fra + ((size_t)n * DD + c) * LL;
    const float* wr = cw + (d * DD + c) * KW;
    #pragma unroll
    for (int j = 0; j < KW; ++j) {
      int p = l + (j - 2) * dil;
      if (p >= 0 && p < LL) acc += wr[j] * base[p];
    }
  }
  s[((size_t)n * DD + d) * LL + l] = acc;
}

// ---------------- window attention (K=5) + residual add into s ----------------
__global__ void attn_kernel(const float* __restrict__ xq, const float* __restrict__ xk,
                            const float* __restrict__ xv, float* __restrict__ s, int dil) {
  const int l = blockIdx.x * blockDim.x + threadIdx.x;
  const int n = blockIdx.y;
  const float LOG_VALID = 9.9999950e-7f;   // log(1 + 1e-6)
  const float LOG_INVALID = -13.815511f;   // log(1e-6)
  float dot[KW]; bool valid[KW]; int off[KW];
  #pragma unroll
  for (int j = 0; j < KW; ++j) {
    off[j] = (j - 2) * dil;
    int p = l + off[j];
    valid[j] = (p >= 0 && p < LL);
    dot[j] = 0.f;
  }
  const float* qp = xq + (size_t)n * DD * LL + l;
  const float* kp = xk + (size_t)n * DD * LL + l;
  for (int d = 0; d < DD; ++d) {
    float qv = qp[(size_t)d * LL];
    #pragma unroll
    for (int j = 0; j < KW; ++j)
      if (valid[j]) dot[j] += qv * kp[(size_t)d * LL + off[j]];
  }
  float att[KW]; float mx = -1e30f;
  #pragma unroll
  for (int j = 0; j < KW; ++j) {
    att[j] = dot[j] * 0.125f + (valid[j] ? LOG_VALID : LOG_INVALID);
    mx = fmaxf(mx, att[j]);
  }
  float ssum = 0.f;
  #pragma unroll
  for (int j = 0; j < KW; ++j) { att[j] = expf(att[j] - mx); ssum += att[j]; }
  const float inv = 1.0f / ssum;
  #pragma unroll
  for (int j = 0; j < KW; ++j) att[j] = valid[j] ? att[j] * inv : 0.0f;
  const float* vp = xv + (size_t)n * DD * LL + l;
  float* sp = s + (size_t)n * DD * LL + l;
  for (int d = 0; d < DD; ++d) {
    float r = 0.f;
    #pragma unroll
    for (int j = 0; j < KW; ++j)
      if (valid[j]) r += att[j] * vp[(size_t)d * LL + off[j]];
    sp[(size_t)d * LL] += r;
  }
}

// ---------------- instance norm over L per (n,c) ----------------
__global__ void inorm_kernel(const float* __restrict__ s, float* __restrict__ h) {
  __shared__ float red[512];
  const int tid = threadIdx.x;
  const int nc = blockIdx.x;
  const float* src = s + (size_t)nc * LL;
  float sum = 0.f, sq = 0.f;
  for (int i = tid; i < LL; i += 256) { float v = src[i]; sum += v; sq += v * v; }
  red[tid] = sum; red[256 + tid] = sq;
  __syncthreads();
  for (int st = 128; st > 0; st >>= 1) {
    if (tid < st) { red[tid] += red[tid + st]; red[256 + tid] += red[256 + tid + st]; }
    __syncthreads();
  }
  const float mean = red[0] / (float)LL;
  const float var = red[256] / (float)LL - mean * mean;
  const float inv = rsqrtf(var + 1e-5f);
  for (int i = tid; i < LL; i += 256)
    h[(size_t)nc * LL + i] = (src[i] - mean) * inv;
}

extern "C" void kernel_launch(void* const* d_in, const int* in_sizes, int n_in,
                              void* d_out, int out_size, void* d_ws, size_t ws_size,
                              hipStream_t stream) {
  (void)in_sizes; (void)n_in; (void)out_size; (void)ws_size;
  const float* x    = (const float*)d_in[0];
  const int*   t    = (const int*)d_in[1];
  const float* ev   = (const float*)d_in[2];
  const float* tw1  = (const float*)d_in[3];
  const float* tb1  = (const float*)d_in[4];
  const float* tw2  = (const float*)d_in[5];
  const float* tb2  = (const float*)d_in[6];
  const float* tpw  = (const float*)d_in[7];
  const float* tpb  = (const float*)d_in[8];
  const float* ciw  = (const float*)d_in[9];
  const float* cib  = (const float*)d_in[10];
  const float* cow  = (const float*)d_in[11];
  const float* cob  = (const float*)d_in[12];
  const float* lcw  = (const float*)d_in[13];
  const float* lcb  = (const float*)d_in[14];
  const float* qw   = (const float*)d_in[15];
  const float* qb   = (const float*)d_in[16];
  const float* kw   = (const float*)d_in[17];
  const float* kb   = (const float*)d_in[18];
  const float* vw   = (const float*)d_in[19];
  const float* vb   = (const float*)d_in[20];
  const float* f1w  = (const float*)d_in[21];
  const float* f1b  = (const float*)d_in[22];
  const float* f2w  = (const float*)d_in[23];
  const float* f2b  = (const float*)d_in[24];
  float* out = (float*)d_out;
  float* ws = (float*)d_ws;

  const size_t NBL = (size_t)BB * DD * LL;
  float* fra  = ws;
  float* xq   = ws + NBL;
  float* xk   = ws + 2 * NBL;
  float* xv   = ws + 3 * NBL;
  float* sbuf = ws + 4 * NBL;
  float* hbuf = ws + 5 * NBL;
  float* ubuf = ws + 6 * NBL;
  float* tvec = ws + 7 * NBL;

  time_embed_kernel<<<BB, TEMBD, 0, stream>>>(t, tw1, tb1, tw2, tb2, tpw, tpb, tvec);
  conv_in_kernel<<<dim3(LL / 256, DD, BB), 256, 0, stream>>>(ciw, cib, ev, tvec, fra);

  for (int i = 0; i < NLAYERS; ++i) {
    const int dil = 1 << i;
    qkproj_kernel<<<dim3(LL / 256, BB), 256, 0, stream>>>(
        qw + (size_t)i * DD * CINQ, qb + i * DD,
        kw + (size_t)i * DD * CINQ, kb + i * DD, fra, x, xq, xk);
    gemm64_kernel<4, false, false><<<dim3(LL / 128, BB), 256, 0, stream>>>(
        vw + (size_t)i * DD * DD, vb + i * DD, fra, nullptr, xv);
    dilconv_kernel<<<dim3(LL / 256, DD, BB), 256, 0, stream>>>(
        lcw + (size_t)i * DD * DD * KW, lcb + i * DD, fra, sbuf, dil);
    attn_kernel<<<dim3(LL / 256, BB), 256, 0, stream>>>(xq, xk, xv, sbuf, dil);
    inorm_kernel<<<BB * DD, 256, 0, stream>>>(sbuf, hbuf);
    gemm64_kernel<4, true, false><<<dim3(LL / 128, BB), 256, 0, stream>>>(
        f1w + (size_t)i * DD * DD, f1b + i * DD, hbuf, nullptr, ubuf);
    gemm64_kernel<4, false, true><<<dim3(LL / 128, BB), 256, 0, stream>>>(
        f2w + (size_t)i * DD * DD, f2b + i * DD, ubuf, fra, fra);
  }
  gemm64_kernel<3, false, false><<<dim3(LL / 128, BB), 256, 0, stream>>>(
      cow, cob, fra, nullptr, out);
}